// SoftGroup_61151744360958
// MI455X (gfx1250) — compile-verified
//
#include <hip/hip_runtime.h>

typedef __attribute__((ext_vector_type(2))) float v2f;
typedef __attribute__((ext_vector_type(8))) float v8f;

#define KNN 16
#define GEO2 (0.06f * 0.06f)
#define COL2 (0.1f * 0.1f)
#define CTILE 4   // column tiles (of 16) per barrier round = 64 columns

// ---------------- kernel 1: squared norms ----------------
__global__ void sqnorm_kernel(const float* __restrict__ coords,
                              float* __restrict__ sq, int n) {
    int i = blockIdx.x * blockDim.x + threadIdx.x;
    if (i < n) {
        float x = coords[3 * i], y = coords[3 * i + 1], z = coords[3 * i + 2];
        sq[i] = x * x + y * y + z * z;
    }
}

// ---------------- kernel 2: KNN via f32 WMMA Gram tiles + per-row top-16 ----
// block = 256 threads = 8 waves; each wave owns a 16-row tile and sweeps the
// 8192 columns in 64-column rounds: 4 back-to-back V_WMMA_F32_16X16X4_F32
// (K=3 padded with 0), one LDS bounce + one barrier pair per round.
__global__ __launch_bounds__(256) void knn_kernel(
        const float* __restrict__ coords, const float* __restrict__ sq,
        int* __restrict__ knn_idx, float* __restrict__ knn_d2, int n) {
    __shared__ float s_tile[8][16][16 * CTILE];   // 32 KB
    __shared__ float s_topd[8][16][KNN];          //  8 KB
    __shared__ int   s_topi[8][16][KNN];          //  8 KB

    const int lane = threadIdx.x & 31;
    const int w    = threadIdx.x >> 5;
    const int rowbase = blockIdx.x * 128 + w * 16;
    const int l  = lane & 15;
    const int hi = lane >> 4;

    // A 16x4 f32 tile: VGPR0={K0 lanes0-15, K2 lanes16-31}, VGPR1={K1, K3}
    v2f a;
    if (hi == 0) { a.x = coords[3 * (rowbase + l) + 0]; a.y = coords[3 * (rowbase + l) + 1]; }
    else         { a.x = coords[3 * (rowbase + l) + 2]; a.y = 0.0f; }

    float sqr[8];
#pragma unroll
    for (int r = 0; r < 8; ++r) sqr[r] = sq[rowbase + r + hi * 8];

    if (lane < 16) {
#pragma unroll
        for (int t = 0; t < KNN; ++t) { s_topd[w][lane][t] = 3.0e38f; s_topi[w][lane][t] = 0; }
    }
    float curmax = 3.0e38f;
    int   curarg = 0;
    __syncthreads();

    for (int cb = 0; cb < n; cb += 16 * CTILE) {
        if (cb + 16 * CTILE < n)  // prefetch next round's B-operand stream
            __builtin_prefetch(&coords[3 * (cb + 16 * CTILE + lane)], 0, 1);

#pragma unroll
        for (int t = 0; t < CTILE; ++t) {
            const int base = cb + 16 * t;
            // B 4x16 tile: same per-lane striping as A (N in lanes)
            v2f b;
            if (hi == 0) { b.x = coords[3 * (base + l) + 0]; b.y = coords[3 * (base + l) + 1]; }
            else         { b.x = coords[3 * (base + l) + 2]; b.y = 0.0f; }
            float sqc = sq[base + l];

            v8f c = {};
            c = __builtin_amdgcn_wmma_f32_16x16x4_f32(
                    /*neg_a=*/false, a, /*neg_b=*/false, b,
                    /*c_mod=*/(short)0, c, /*reuse_a=*/false, /*reuse_b=*/false);

            // D layout: VGPR r -> row rowbase + r + 8*hi, col = base + (lane&15)
#pragma unroll
            for (int r = 0; r < 8; ++r) {
                float d2 = sqr[r] + sqc - 2.0f * c[r];
                s_tile[w][r + hi * 8][t * 16 + l] = d2;
            }
        }
        __syncthreads();

        // lanes 0-15 each own one row's top-16 (kept in LDS: dynamic index)
        if (lane < 16) {
            for (int j = 0; j < 16 * CTILE; ++j) {
                float d2 = s_tile[w][lane][j];
                if (d2 < curmax) {
                    s_topd[w][lane][curarg] = d2;
                    s_topi[w][lane][curarg] = cb + j;
                    float m = -3.0e38f; int am = 0;
#pragma unroll
                    for (int t = 0; t < KNN; ++t) {
                        float v = s_topd[w][lane][t];
                        if (v > m) { m = v; am = t; }
                    }
                    curmax = m; curarg = am;
                }
            }
        }
        __syncthreads();
    }

    if (lane < 16) {
        int row = rowbase + lane;
#pragma unroll
        for (int t = 0; t < KNN; ++t) {
            knn_idx[row * KNN + t] = s_topi[w][lane][t];
            knn_d2 [row * KNN + t] = s_topd[w][lane][t];
        }
    }
}

// ---------------- kernel 3: valid-edge neighbor lists (-1 = invalid) -------
__global__ void edge_kernel(const float* __restrict__ rgb,
                            const int* __restrict__ knn_idx,
                            const float* __restrict__ knn_d2,
                            int* __restrict__ nbr, int n) {
    int t = blockIdx.x * blockDim.x + threadIdx.x;
    if (t >= n * KNN) return;
    int u = t / KNN;
    int v = knn_idx[t];
    float d2 = knn_d2[t];
    float dr = rgb[3 * v + 0] - rgb[3 * u + 0];
    float dg = rgb[3 * v + 1] - rgb[3 * u + 1];
    float db = rgb[3 * v + 2] - rgb[3 * u + 2];
    float cd2 = dr * dr + dg * dg + db * db;
    nbr[t] = (d2 < GEO2 && cd2 < COL2) ? v : -1;
}

// ---------------- kernel 4: zero the packed (level,pivot) keys -------------
__global__ void initkey_kernel(unsigned* __restrict__ key, int n) {
    int i = blockIdx.x * blockDim.x + threadIdx.x;
    if (i < n) key[i] = 0u;
}

// ---------------- kernel 5: sparse bitset BFS, one wave per pivot ----------
// key[n] = max over pivots of ((level<<12) | (P-1-p)); level in {4,3,2,1}.
// Max level wins; on ties the smallest pivot index wins (== jnp.argmax).
__global__ __launch_bounds__(32) void bfs_kernel(
        const int* __restrict__ knn_idx, const int* __restrict__ nbr,
        const int* __restrict__ pivot_inds, unsigned* __restrict__ key,
        int n, int p_total) {
    __shared__ unsigned vis[256], fr[256], nx[256];  // 8192 nodes / 32
    const int p = blockIdx.x;
    const int lane = threadIdx.x;
    const int nw = n >> 5;
    const int pivot = pivot_inds[p];
    const unsigned prank = (unsigned)(p_total - 1 - p);

    for (int i = lane; i < nw; i += 32) { vis[i] = 0u; fr[i] = 0u; }
    __syncthreads();

    // initial frontier: unfiltered 16-NN of the pivot (includes pivot itself)
    if (lane < KNN) {
        int v = knn_idx[pivot * KNN + lane];
        atomicOr(&fr[v >> 5], 1u << (v & 31));
    }
    __syncthreads();

    for (int step = 0; step < 4; ++step) {
        unsigned pack = (((unsigned)(4 - step)) << 12) | prank;
        for (int i = lane; i < nw; i += 32) nx[i] = 0u;
        __syncthreads();
        for (int i = lane; i < nw; i += 32) {
            unsigned newly = fr[i] & ~vis[i];
            vis[i] |= newly;
            if (newly) {
                atomicOr(&nx[i], newly);  // "frontier = nbrs | newly"
                unsigned m = newly;
                while (m) {
                    int b = __builtin_ctz(m);
                    m &= m - 1;
                    int u = (i << 5) + b;
                    atomicMax(&key[u], pack);
#pragma unroll
                    for (int k2 = 0; k2 < KNN; ++k2) {
                        int v = nbr[u * KNN + k2];
                        if (v >= 0) atomicOr(&nx[v >> 5], 1u << (v & 31));
                    }
                }
            }
        }
        __syncthreads();
        for (int i = lane; i < nw; i += 32) fr[i] = nx[i];
        __syncthreads();
    }
}

// ---------------- kernel 6: decode keys -> (pseudo_labels, max_dist) -------
__global__ void final_kernel(const unsigned* __restrict__ key,
                             const int* __restrict__ sem_labels,
                             const int* __restrict__ pivot_inds,
                             float* __restrict__ out, int n, int p_total) {
    int i = blockIdx.x * blockDim.x + threadIdx.x;
    if (i >= n) return;
    unsigned k = key[i];
    float lab, md;
    if (k == 0u) {
        lab = -100.0f; md = 0.0f;
    } else {
        int level = (int)(k >> 12);
        int p = p_total - 1 - (int)(k & 0xFFFu);
        lab = (float)sem_labels[pivot_inds[p]];
        md  = (float)level;
    }
    out[i]     = lab;
    out[n + i] = md;
}

extern "C" void kernel_launch(void* const* d_in, const int* in_sizes, int n_in,
                              void* d_out, int out_size, void* d_ws, size_t ws_size,
                              hipStream_t stream) {
    const float* coords = (const float*)d_in[0];
    const float* rgb    = (const float*)d_in[1];
    const int*   sem    = (const int*)d_in[2];
    const int*   piv    = (const int*)d_in[3];
    const int n = in_sizes[0] / 3;   // 8192
    const int p = in_sizes[3];       // 512

    char* ws = (char*)d_ws;
    float*    sq      = (float*)ws;    ws += (size_t)n * sizeof(float);
    int*      knn_idx = (int*)ws;      ws += (size_t)n * KNN * sizeof(int);
    float*    knn_d2  = (float*)ws;    ws += (size_t)n * KNN * sizeof(float);
    int*      nbr     = (int*)ws;      ws += (size_t)n * KNN * sizeof(int);
    unsigned* key     = (unsigned*)ws; ws += (size_t)n * sizeof(unsigned);

    sqnorm_kernel <<<(n + 255) / 256, 256, 0, stream>>>(coords, sq, n);
    knn_kernel    <<<n / 128,          256, 0, stream>>>(coords, sq, knn_idx, knn_d2, n);
    edge_kernel   <<<(n * KNN + 255) / 256, 256, 0, stream>>>(rgb, knn_idx, knn_d2, nbr, n);
    initkey_kernel<<<(n + 255) / 256, 256, 0, stream>>>(key, n);
    bfs_kernel    <<<p, 32, 0, stream>>>(knn_idx, nbr, piv, key, n, p);
    final_kernel  <<<(n + 255) / 256, 256, 0, stream>>>(key, sem, piv, (float*)d_out, n, p);
}